// GCN_35476429865194
// MI455X (gfx1250) — compile-verified
//
#include <hip/hip_runtime.h>
#include <hip/hip_bf16.h>

typedef float v2f __attribute__((ext_vector_type(2)));
typedef float v8f __attribute__((ext_vector_type(8)));

#define IN_FEATS 128
#define HIDDEN   64

// ---------------------------------------------------------------------------
// Degree: deg[i] starts at 1.0 (self loop), then +1 per incoming edge.
// ---------------------------------------------------------------------------
__global__ void k_init_deg(float* __restrict__ deg, int n) {
    int i = blockIdx.x * blockDim.x + threadIdx.x;
    if (i < n) deg[i] = 1.0f;
}

__global__ void k_accum_deg(const int* __restrict__ dst, int E, float* __restrict__ deg) {
    int e = blockIdx.x * blockDim.x + threadIdx.x;
    if (e < E) atomicAdd(&deg[dst[e]], 1.0f);
}

__global__ void k_make_dinv(const float* __restrict__ deg, float* __restrict__ dinv, int n) {
    int i = blockIdx.x * blockDim.x + threadIdx.x;
    if (i < n) {
        float d = deg[i];
        dinv[i] = (d > 0.0f) ? rsqrtf(d) : 0.0f;
    }
}

// ---------------------------------------------------------------------------
// GEMM1: H[N x 64] = X[N x 128] @ W1[128 x 64], fp32 WMMA 16x16x4.
// One wave per 16x16 output tile. W1 staged in LDS (32 KB).
// A frag (16x4): lane L -> row L&15, K = 2*(L>>4)+{0,1}
// B frag (4x16): lane L -> col L&15, K rows 2*(L>>4)+{0,1} in vgpr {0,1}
// C/D (16x16):   vgpr v -> row v + 8*(L>>4), col L&15
// ---------------------------------------------------------------------------
__global__ void __launch_bounds__(256) k_gemm1_wmma(const float* __restrict__ X,
                                                    const float* __restrict__ W1,
                                                    float* __restrict__ H,
                                                    int nTiles) {
    __shared__ float Wlds[IN_FEATS * HIDDEN];  // 32 KB
    for (int i = threadIdx.x; i < IN_FEATS * HIDDEN; i += blockDim.x)
        Wlds[i] = W1[i];
    __syncthreads();

    const int waveId = (int)((blockIdx.x * blockDim.x + threadIdx.x) >> 5);
    const int lane   = threadIdx.x & 31;
    if (waveId >= nTiles) return;          // wave-uniform guard; EXEC stays all-ones

    const int nTilesN = HIDDEN / 16;       // 4
    const int mt = waveId / nTilesN;
    const int nt = waveId % nTilesN;

    const int row = mt * 16 + (lane & 15);
    const int n   = nt * 16 + (lane & 15);
    const int kq  = (lane >> 4) << 1;      // 0 or 2

    const float* __restrict__ xrow = X + (size_t)row * IN_FEATS;

    v8f c = {};
    #pragma unroll 8
    for (int k = 0; k < IN_FEATS; k += 4) {
        v2f a, b;
        a.x = xrow[k + kq];
        a.y = xrow[k + kq + 1];
        b.x = Wlds[(k + kq)     * HIDDEN + n];
        b.y = Wlds[(k + kq + 1) * HIDDEN + n];
        c = __builtin_amdgcn_wmma_f32_16x16x4_f32(
                /*neg_a=*/false, a, /*neg_b=*/false, b,
                /*c_mod=*/(short)0, c, /*reuse_a=*/false, /*reuse_b=*/false);
    }

    const int rbase = mt * 16 + ((lane >> 4) << 3);  // +8 for upper half-wave
    #pragma unroll
    for (int v = 0; v < 8; ++v)
        H[(size_t)(rbase + v) * HIDDEN + n] = c[v];
}

// ---------------------------------------------------------------------------
// Layer-1 aggregation. Accumulator seeded with the self-loop message
// (norm = dinv^2), then 1 wave per edge scatter-adds H[src]*norm into A1[dst].
// H and A1 are 12.8 MB each -> resident in the 192 MB L2; atomics are L2-local.
// ---------------------------------------------------------------------------
__global__ void k_self_init1(const float* __restrict__ H, const float* __restrict__ dinv,
                             float* __restrict__ A1, int total /* N*64 */) {
    int i = blockIdx.x * blockDim.x + threadIdx.x;
    if (i < total) {
        float d = dinv[i >> 6];
        A1[i] = H[i] * d * d;
    }
}

__global__ void __launch_bounds__(256) k_agg1(const int* __restrict__ src,
                                              const int* __restrict__ dst,
                                              const float* __restrict__ dinv,
                                              const float* __restrict__ H,
                                              float* __restrict__ A1, int E) {
    const int wave = (int)((blockIdx.x * blockDim.x + threadIdx.x) >> 5);
    const int lane = threadIdx.x & 31;
    if (wave >= E) return;
    const int s = src[wave];
    const int d = dst[wave];
    const float w = dinv[s] * dinv[d];
    const float* __restrict__ hs = H  + (size_t)s * HIDDEN;
    float* __restrict__       ad = A1 + (size_t)d * HIDDEN;
    atomicAdd(&ad[lane],      hs[lane]      * w);
    atomicAdd(&ad[lane + 32], hs[lane + 32] * w);
}

// ---------------------------------------------------------------------------
// Fused bias + ReLU + tiny GEMM: H2[N x 2] = relu(A1 + b1) @ W2[64 x 2].
// 128 FLOP/node -> plain VALU, one thread per node.
// ---------------------------------------------------------------------------
__global__ void k_dense2(const float* __restrict__ A1, const float* __restrict__ b1,
                         const float* __restrict__ W2, float* __restrict__ H2, int n) {
    int i = blockIdx.x * blockDim.x + threadIdx.x;
    if (i >= n) return;
    const float* __restrict__ a = A1 + (size_t)i * HIDDEN;
    float acc0 = 0.0f, acc1 = 0.0f;
    #pragma unroll 16
    for (int f = 0; f < HIDDEN; ++f) {
        float v = a[f] + b1[f];
        v = v > 0.0f ? v : 0.0f;
        acc0 = fmaf(v, W2[f * 2 + 0], acc0);
        acc1 = fmaf(v, W2[f * 2 + 1], acc1);
    }
    H2[i * 2 + 0] = acc0;
    H2[i * 2 + 1] = acc1;
}

// ---------------------------------------------------------------------------
// Layer-2 aggregation (2 features): seed with self-loop + bias, then
// one thread per edge.
// ---------------------------------------------------------------------------
__global__ void k_self_init2(const float* __restrict__ H2, const float* __restrict__ dinv,
                             const float* __restrict__ b2, float* __restrict__ out, int n) {
    int i = blockIdx.x * blockDim.x + threadIdx.x;
    if (i < n) {
        float d = dinv[i];
        float w = d * d;
        out[i * 2 + 0] = H2[i * 2 + 0] * w + b2[0];
        out[i * 2 + 1] = H2[i * 2 + 1] * w + b2[1];
    }
}

__global__ void k_agg2(const int* __restrict__ src, const int* __restrict__ dst,
                       const float* __restrict__ dinv, const float* __restrict__ H2,
                       float* __restrict__ out, int E) {
    int e = blockIdx.x * blockDim.x + threadIdx.x;
    if (e >= E) return;
    const int s = src[e];
    const int d = dst[e];
    const float w = dinv[s] * dinv[d];
    atomicAdd(&out[d * 2 + 0], H2[s * 2 + 0] * w);
    atomicAdd(&out[d * 2 + 1], H2[s * 2 + 1] * w);
}

// ---------------------------------------------------------------------------
// Launcher
// in: 0=x[N*128] 1=W1[128*64] 2=b1[64] 3=W2[64*2] 4=b2[2] 5=edge_index[2*E]
// ---------------------------------------------------------------------------
extern "C" void kernel_launch(void* const* d_in, const int* in_sizes, int n_in,
                              void* d_out, int out_size, void* d_ws, size_t ws_size,
                              hipStream_t stream) {
    const float* x  = (const float*)d_in[0];
    const float* W1 = (const float*)d_in[1];
    const float* b1 = (const float*)d_in[2];
    const float* W2 = (const float*)d_in[3];
    const float* b2 = (const float*)d_in[4];
    const int*   ei = (const int*)d_in[5];

    const int N = in_sizes[0] / IN_FEATS;   // 50000
    const int E = in_sizes[5] / 2;          // 1600000
    const int* src = ei;
    const int* dst = ei + E;

    float* ws   = (float*)d_ws;
    float* deg  = ws;                         // N
    float* dinv = deg  + N;                   // N
    float* H    = dinv + N;                   // N*64
    float* A1   = H    + (size_t)N * HIDDEN;  // N*64
    float* H2   = A1   + (size_t)N * HIDDEN;  // N*2
    float* out  = (float*)d_out;              // N*2

    const int T = 256;

    // degree + norm
    k_init_deg <<<(N + T - 1) / T, T, 0, stream>>>(deg, N);
    k_accum_deg<<<(E + T - 1) / T, T, 0, stream>>>(dst, E, deg);
    k_make_dinv<<<(N + T - 1) / T, T, 0, stream>>>(deg, dinv, N);

    // layer 1: WMMA GEMM then edge aggregation
    const int nTiles = (N / 16) * (HIDDEN / 16);          // 12500 waves
    const int gemmBlocks = (nTiles + (T / 32) - 1) / (T / 32);
    k_gemm1_wmma<<<gemmBlocks, T, 0, stream>>>(x, W1, H, nTiles);

    k_self_init1<<<((size_t)N * HIDDEN + T - 1) / T, T, 0, stream>>>(H, dinv, A1, N * HIDDEN);
    {
        const long long waves = (long long)E;
        const long long blocks = (waves * 32 + T - 1) / T;
        k_agg1<<<(int)blocks, T, 0, stream>>>(src, dst, dinv, H, A1, E);
    }

    // layer 2
    k_dense2    <<<(N + T - 1) / T, T, 0, stream>>>(A1, b1, W2, H2, N);
    k_self_init2<<<(N + T - 1) / T, T, 0, stream>>>(H2, dinv, b2, out, N);
    k_agg2      <<<(E + T - 1) / T, T, 0, stream>>>(src, dst, dinv, H2, out, E);
}